// SSIM3D_44349832298988
// MI455X (gfx1250) — compile-verified
//
#include <hip/hip_runtime.h>

typedef float v2f __attribute__((ext_vector_type(2)));
typedef float v4f __attribute__((ext_vector_type(4)));
typedef float v8f __attribute__((ext_vector_type(8)));
typedef unsigned int u32x4 __attribute__((ext_vector_type(4)));
typedef int i32x4 __attribute__((ext_vector_type(4)));
typedef int i32x8 __attribute__((ext_vector_type(8)));

#define D_DIM  160
#define H_DIM  192
#define W_DIM  192
#define TS     16                 // output tile edge (H,W)
#define HALO   5                  // 11-tap window
#define TIN    26                 // TS + 2*HALO
#define DCHUNK 8                  // output D slices per block
#define NSLICE (DCHUNK + 2*HALO)  // 18 input slices per block
#define TP     33                 // LDS tile pitch (odd -> conflict-free column walks)

#if defined(__has_builtin)
#if __has_builtin(__builtin_amdgcn_tensor_load_to_lds) && __has_builtin(__builtin_amdgcn_s_wait_tensorcnt)
#define USE_TDM 1
#endif
#endif

// normalized 11-tap gaussian, sigma=1.5 (precomputed in double, rounded to f32)
__device__ __constant__ float GTAB[11] = {
    0.00102840f, 0.00759877f, 0.03600077f, 0.10936153f, 0.21300535f,
    0.26601198f, 0.21300535f, 0.10936153f, 0.03600077f, 0.00759877f, 0.00102840f
};

// banded Gaussian matrix entry: G[k][n] = g[k-n] if 0 <= k-n <= 10 else 0
__device__ __forceinline__ float bandv(int k, int n, const float* gs) {
    int t = k - n;
    return (t >= 0 && t <= 10) ? gs[t] : 0.0f;
}

__device__ __forceinline__ v2f chanv2(int c, v2f a, v2f b) {
    switch (c) {
        case 0:  return a;
        case 1:  return b;
        case 2:  return a * a;
        case 3:  return b * b;
        default: return a * b;
    }
}

#ifdef USE_TDM
// Issue one TDM 2D tile load: 33x26 f32 tile at gptr (row pitch W_DIM elems)
// into LDS at byte offset lds_off (row pitch TP=33 floats, no padding needed).
// tensor_dim0/1 = 192 gives automatic zero-fill past the right/bottom edges.
__device__ __forceinline__ void tdm_load_tile(const float* gptr, unsigned lds_off)
{
    unsigned long long ga = (unsigned long long)(uintptr_t)gptr;
    u32x4 g0;
    g0.x = 1u;                                                  // count=1 valid user descriptor
    g0.y = lds_off;                                             // lds_addr (bytes)
    g0.z = (unsigned)ga;                                        // global_addr[31:0]
    g0.w = (unsigned)((ga >> 32) & 0x01FFFFFFull) | (2u << 30); // global_addr[56:32] | type=2
    i32x8 g1;
    g1[0] = (int)(2u << 16);              // wg_mask=0, data_size=4B, no pad/iterate
    g1[1] = (int)((unsigned)W_DIM << 16); // tensor_dim0 = 192 (bits 63:48)
    g1[2] = (int)((unsigned)H_DIM << 16); // dim0 hi=0 | tensor_dim1 = 192 (bits 95:80)
    g1[3] = (int)(33u << 16);             // dim1 hi=0 | tile_dim0 = 33
    g1[4] = TIN;                          // tile_dim1 = 26, tile_dim2 = 0 (2D)
    g1[5] = W_DIM;                        // tensor_dim0_stride low32 = 192
    g1[6] = (int)((unsigned)(H_DIM * W_DIM) << 16); // stride0 hi=0 | stride1 low16
    g1[7] = 0;                            // stride1 hi
    i32x4 gz4 = { 0, 0, 0, 0 };
    i32x8 gz8 = { 0, 0, 0, 0, 0, 0, 0, 0 };
    // 6-arg toolchain variant: (g0, g1, g2, g3, g4, cpol)
    __builtin_amdgcn_tensor_load_to_lds(g0, g1, gz4, gz4, gz8, 0);
}
#endif

__launch_bounds__(32)
__global__ void ssim3d_fused(const float* __restrict__ img1,
                             const float* __restrict__ img2,
                             float* __restrict__ partials)
{
    __shared__ float t1[TIN * TP];        // img1 halo tile (one D slice)
    __shared__ float t2[TIN * TP];        // img2 halo tile
    __shared__ float w1s[32 * TS];        // W-blurred intermediate (rows 26..31 carry finite
                                          // garbage multiplied by exact-zero band coeffs)
    __shared__ float ring[11 * 5 * 256];  // [slot][chan][lane*8+v] 2D-blurred stats ring
    __shared__ float gs[16];              // gaussian taps
    __shared__ float red[32];

    const int lane  = threadIdx.x;        // one wave32 per block
    const int w0    = blockIdx.x * TS;
    const int h0    = blockIdx.y * TS;
    const int d0    = blockIdx.z * DCHUNK;
    const int khalf = lane >> 4;          // f32 WMMA: lanes 0-15 hold K0/K1, 16-31 hold K2/K3
    const int nlo   = lane & 15;

    if (lane < 11) gs[lane] = GTAB[lane];
    __syncthreads();

    // ---- hoist the banded-Gaussian fragments (loop invariant per lane) ----
    // Same values serve as B of the W pass and A of the H pass.
    v2f Bg[8];
    #pragma unroll
    for (int kc = 0; kc < 8; ++kc) {
        const int k = 4 * kc + 2 * khalf;
        Bg[kc].x = bandv(k,     nlo, gs);
        Bg[kc].y = bandv(k + 1, nlo, gs);
    }

#ifdef USE_TDM
    const bool     interior = (w0 > 0) && (h0 > 0);   // right/bottom handled by tensor_dim OOB-zero
    const unsigned lds_t1   = (unsigned)(uintptr_t)(void*)t1;
    const unsigned lds_t2   = (unsigned)(uintptr_t)(void*)t2;
#endif

    float part = 0.f;

    for (int ls = 0; ls < NSLICE; ++ls) {
        const int  s    = d0 - HALO + ls;
        const bool sval = (s >= 0) && (s < D_DIM);
        const long soff = (long)s * (H_DIM * W_DIM);

        // ---- halo tile load: TDM DMA for interior tiles, scalar path otherwise ----
        bool tdm_done = false;
#ifdef USE_TDM
        if (sval && interior) {
            const long tile_off = soff + (long)(h0 - HALO) * W_DIM + (w0 - HALO);
            tdm_load_tile(img1 + tile_off, lds_t1);
            tdm_load_tile(img2 + tile_off, lds_t2);
            __builtin_amdgcn_s_wait_tensorcnt(0);
            tdm_done = true;
        }
#endif
        if (!tdm_done) {
            for (int y = 0; y < TIN; ++y) {
                const int x  = lane;                  // cols 0..31 (col 32 never read)
                const int gy = h0 - HALO + y;
                const int gx = w0 - HALO + x;
                float a = 0.f, b = 0.f;
                if (sval && x < TIN && gy >= 0 && gy < H_DIM && gx >= 0 && gx < W_DIM) {
                    long off = soff + (long)gy * W_DIM + gx;
                    a = img1[off];
                    b = img2[off];
                }
                t1[y * TP + x] = a;
                t2[y * TP + x] = b;
            }
        }
        // prefetch a row of the next slice (global_prefetch_b8)
        if ((unsigned)(s + 1) < (unsigned)D_DIM && lane < TIN) {
            int gy = h0 - HALO + lane; gy = gy < 0 ? 0 : (gy >= H_DIM ? H_DIM - 1 : gy);
            int gx = w0 - HALO;        gx = gx < 0 ? 0 : gx;
            long off = ((long)(s + 1) * H_DIM + gy) * W_DIM + gx;
            __builtin_prefetch(img1 + off, 0, 3);
            __builtin_prefetch(img2 + off, 0, 3);
        }
        __syncthreads();

        // ---- hoist raw A-tile values once per slice, reused by all 5 channels ----
        const int y0 = nlo;                           // row group 0
        int y1 = nlo + 16; if (y1 > TIN - 1) y1 = TIN - 1;   // row group 1 (rows >25 discarded)
        v2f ta0[8], tb0[8], ta1[8], tb1[8];
        #pragma unroll
        for (int kc = 0; kc < 8; ++kc) {
            const int k = 4 * kc + 2 * khalf;
            ta0[kc].x = t1[y0 * TP + k]; ta0[kc].y = t1[y0 * TP + k + 1];
            tb0[kc].x = t2[y0 * TP + k]; tb0[kc].y = t2[y0 * TP + k + 1];
            ta1[kc].x = t1[y1 * TP + k]; ta1[kc].y = t1[y1 * TP + k + 1];
            tb1[kc].x = t2[y1 * TP + k]; tb1[kc].y = t2[y1 * TP + k + 1];
        }

        const int slot = (ls < 11) ? ls : (ls - 11);

        #pragma unroll
        for (int c = 0; c < 5; ++c) {
            // ---- W pass: out = X(16x32) * G(32x16), 8 chained K=4 f32 WMMAs per row group ----
            {
                v8f acc = {};
                #pragma unroll
                for (int kc = 0; kc < 8; ++kc)
                    acc = __builtin_amdgcn_wmma_f32_16x16x4_f32(
                              false, chanv2(c, ta0[kc], tb0[kc]), false, Bg[kc],
                              (short)0, acc, false, false);
                #pragma unroll
                for (int v = 0; v < 8; ++v)           // rows 0..15
                    w1s[(v + 8 * khalf) * TS + nlo] = acc[v];
            }
            {
                v8f acc = {};
                #pragma unroll
                for (int kc = 0; kc < 8; ++kc)
                    acc = __builtin_amdgcn_wmma_f32_16x16x4_f32(
                              false, chanv2(c, ta1[kc], tb1[kc]), false, Bg[kc],
                              (short)0, acc, false, false);
                // rows 16..31 stored unconditionally: rows >= 26 hold finite values that the
                // H pass multiplies by exact-zero band coefficients (k - m >= 11) -> contribute 0
                #pragma unroll
                for (int v = 0; v < 8; ++v)
                    w1s[(16 + v + 8 * khalf) * TS + nlo] = acc[v];
            }
            __syncthreads();

            // ---- H pass: out = G^T(16x32 banded) * Y(32x16) ----
            v8f acc2 = {};
            #pragma unroll
            for (int kc = 0; kc < 8; ++kc) {
                const int k = 4 * kc + 2 * khalf;
                v2f B;
                B.x = w1s[k * TS + nlo];
                B.y = w1s[(k + 1) * TS + nlo];
                acc2 = __builtin_amdgcn_wmma_f32_16x16x4_f32(
                           false, Bg[kc], false, B, (short)0, acc2, false, false);
            }
            float* rg = &ring[(slot * 5 + c) * 256 + lane * 8];
            v4f lo = { acc2[0], acc2[1], acc2[2], acc2[3] };
            v4f hi = { acc2[4], acc2[5], acc2[6], acc2[7] };
            *(v4f*)rg       = lo;
            *(v4f*)(rg + 4) = hi;
            __syncthreads();
        }

        // ---- D pass + SSIM once the 11-slice ring is warm ----
        if (ls >= 2 * HALO) {
            const int base = ls - 2 * HALO;           // = d - d0
            float st[5][8];
            #pragma unroll
            for (int c = 0; c < 5; ++c)
                #pragma unroll
                for (int v = 0; v < 8; ++v) st[c][v] = 0.f;
            #pragma unroll
            for (int j = 0; j < 11; ++j) {
                int sl = base + j; if (sl >= 11) sl -= 11;
                const float gj = gs[j];
                #pragma unroll
                for (int c = 0; c < 5; ++c) {
                    const float* rb = &ring[(sl * 5 + c) * 256 + lane * 8];
                    v4f lo = *(const v4f*)rb;
                    v4f hi = *(const v4f*)(rb + 4);
                    st[c][0] += gj * lo.x; st[c][1] += gj * lo.y;
                    st[c][2] += gj * lo.z; st[c][3] += gj * lo.w;
                    st[c][4] += gj * hi.x; st[c][5] += gj * hi.y;
                    st[c][6] += gj * hi.z; st[c][7] += gj * hi.w;
                }
            }
            #pragma unroll
            for (int v = 0; v < 8; ++v) {
                const float mu1 = st[0][v], mu2 = st[1][v];
                const float mu1s = mu1 * mu1, mu2s = mu2 * mu2, mu12 = mu1 * mu2;
                const float s1  = st[2][v] - mu1s;
                const float s2v = st[3][v] - mu2s;
                const float s12 = st[4][v] - mu12;
                const float C1 = 1e-4f, C2 = 9e-4f;
                const float num = (2.f * mu12 + C1) * (2.f * s12 + C2);
                const float den = (mu1s + mu2s + C1) * (s1 + s2v + C2);
                part += num / den;
            }
        }
        __syncthreads();
    }

    // ---- per-block reduction, partial to workspace (deterministic two-stage mean) ----
    red[lane] = part;
    __syncthreads();
    if (lane == 0) {
        float tot = 0.f;
        #pragma unroll
        for (int i = 0; i < 32; ++i) tot += red[i];
        const int bid = (blockIdx.z * gridDim.y + blockIdx.y) * gridDim.x + blockIdx.x;
        partials[bid] = tot;
    }
}

__launch_bounds__(32)
__global__ void ssim3d_reduce(const float* __restrict__ partials,
                              float* __restrict__ out, int n)
{
    __shared__ float red[32];
    float s = 0.f;
    for (int i = threadIdx.x; i < n; i += 32) s += partials[i];
    red[threadIdx.x] = s;
    __syncthreads();
    if (threadIdx.x == 0) {
        float t = 0.f;
        #pragma unroll
        for (int i = 0; i < 32; ++i) t += red[i];
        out[0] = t * (1.0f / (float)((long)D_DIM * H_DIM * W_DIM));
    }
}

extern "C" void kernel_launch(void* const* d_in, const int* in_sizes, int n_in,
                              void* d_out, int out_size, void* d_ws, size_t ws_size,
                              hipStream_t stream)
{
    const float* img1 = (const float*)d_in[0];
    const float* img2 = (const float*)d_in[1];
    float* out      = (float*)d_out;
    float* partials = (float*)d_ws;

    dim3 grid(W_DIM / TS, H_DIM / TS, D_DIM / DCHUNK);   // 12 x 12 x 20 = 2880 blocks
    ssim3d_fused<<<grid, 32, 0, stream>>>(img1, img2, partials);

    const int nblk = (W_DIM / TS) * (H_DIM / TS) * (D_DIM / DCHUNK);
    ssim3d_reduce<<<1, 32, 0, stream>>>(partials, out, nblk);
}